// RESCAL_22282290331969
// MI455X (gfx1250) — compile-verified
//
#include <hip/hip_runtime.h>

// RESCAL scoring on MI455X (gfx1250): score_b = h_b^T * R_b * t_b, fp32.
// Bandwidth/L2-bound (1.07 GB of gathered R per call vs 0.54 GFLOP): rel_w
// (131 MB) + ent_w (51 MB) fit in the 192 MB L2, so the kernel streams R
// tiles from L2. Matvec runs on V_WMMA_F32_16X16X4_F32 (fp32, matching the
// reference precision); B operand holds the t-chunk in column 0 only, built
// via a zero-region LDS trick so the inner loop has uniform control flow.

#define DIMN   128
#define BATCHN 16384
#define ALPHAF 0.001f

typedef __attribute__((ext_vector_type(2))) float v2f;
typedef __attribute__((ext_vector_type(8))) float v8f;

__global__ __launch_bounds__(128) void rescal_score_kernel(
    const int*   __restrict__ h_idx,
    const int*   __restrict__ r_idx,
    const int*   __restrict__ t_idx,
    const float* __restrict__ labels,
    const float* __restrict__ ent_w,
    const float* __restrict__ rel_w,
    float*       __restrict__ scores_out,   // [BATCHN]
    float*       __restrict__ partials)     // [BATCHN][4]: sqerr, sum h^2, sum t^2, sum r^2
{
    __shared__ float sh[DIMN];        // h_b
    __shared__ float sB[2 * DIMN];    // [0..127] = t_b, [128..255] = zeros
    __shared__ float sRed[128];       // reduction scratch
    __shared__ float sP[8];           // per-(wave,half) score partials

    const int b    = blockIdx.x;
    const int tid  = threadIdx.x;      // 0..127 == DIM
    const int lane = tid & 31;
    const int wave = tid >> 5;         // 0..3
    const int mrow = lane & 15;        // row within 16-row tile
    const int hi   = lane >> 4;        // 0: K=0,1   1: K=2,3

    // Gather h and t rows (one float per thread), stage in LDS.
    const float hv = ent_w[(size_t)h_idx[b] * DIMN + tid];
    const float tv = ent_w[(size_t)t_idx[b] * DIMN + tid];
    sh[tid]        = hv;
    sB[tid]        = tv;
    sB[DIMN + tid] = 0.f;              // zero region for inactive B lanes
    float hh = hv * hv;
    float tt = tv * tv;
    __syncthreads();

    const float* __restrict__ Rb = rel_w + (size_t)r_idx[b] * (DIMN * DIMN);

    // Wave w handles rows [32w, 32w+32): two 16-row WMMA tiles.
    const int r0 = 32 * wave + mrow;   // tile 0 row for this lane
    const int r1 = r0 + 16;            // tile 1 row

    // B operand base: lanes 0/16 read t[k..k+1]/t[k+2..k+3]; all other lanes
    // read from the zero region, so column 0 of B is the t-chunk and the rest 0.
    const int bbase = (mrow == 0 ? 0 : DIMN) + 2 * hi;

    v8f acc0 = {};
    v8f acc1 = {};
    v2f rr2  = {};                     // packed sum of R^2

#pragma unroll 8
    for (int s = 0; s < 32; ++s) {
        const int k = 4 * s + 2 * hi;  // this lane's K pair

        v2f Bop = *(const v2f*)(sB + bbase + 4 * s);   // ds_load_b64, uniform flow

        v2f A0 = *(const v2f*)(Rb + (size_t)r0 * DIMN + k);
        rr2 += A0 * A0;                                // v_pk_fma_f32
        acc0 = __builtin_amdgcn_wmma_f32_16x16x4_f32(
            false, A0, false, Bop, (short)0, acc0, false, false);

        v2f A1 = *(const v2f*)(Rb + (size_t)r1 * DIMN + k);
        rr2 += A1 * A1;
        acc1 = __builtin_amdgcn_wmma_f32_16x16x4_f32(
            false, A1, false, Bop, (short)0, acc1, false, false);
    }
    float rr = rr2.x + rr2.y;

    // Column 0 of D: lane 0 holds tr[m0+0..7] in acc[0..7], lane 16 holds tr[m0+8..15].
    if (mrow == 0) {
        const int base = 32 * wave + 8 * hi;
        float part = 0.f;
#pragma unroll
        for (int i = 0; i < 8; ++i) {
            part += sh[base + i]      * acc0[i];
            part += sh[base + 16 + i] * acc1[i];
        }
        sP[wave * 2 + hi] = part;
    }
    __syncthreads();

    // Deterministic tree reductions of hh, tt, rr across the block.
    float vals[3] = {hh, tt, rr};
    float sums[3];
#pragma unroll
    for (int j = 0; j < 3; ++j) {
        sRed[tid] = vals[j];
        __syncthreads();
        for (int off = 64; off > 0; off >>= 1) {
            if (tid < off) sRed[tid] += sRed[tid + off];
            __syncthreads();
        }
        sums[j] = sRed[0];
        __syncthreads();
    }

    if (tid == 0) {
        float score = 0.f;
#pragma unroll
        for (int i = 0; i < 8; ++i) score += sP[i];
        scores_out[b] = score;
        const float d = score - labels[b];
        float* p = partials + (size_t)b * 4;
        p[0] = d * d;
        p[1] = sums[0];
        p[2] = sums[1];
        p[3] = sums[2];
    }
}

__global__ __launch_bounds__(256) void rescal_loss_kernel(
    const float* __restrict__ partials,
    float*       __restrict__ loss_out)
{
    __shared__ float s0[256], s1[256], s2[256], s3[256];
    const int tid = threadIdx.x;
    float a0 = 0.f, a1 = 0.f, a2 = 0.f, a3 = 0.f;
    for (int i = tid; i < BATCHN; i += 256) {
        const float* p = partials + (size_t)i * 4;
        a0 += p[0]; a1 += p[1]; a2 += p[2]; a3 += p[3];
    }
    s0[tid] = a0; s1[tid] = a1; s2[tid] = a2; s3[tid] = a3;
    __syncthreads();
    for (int off = 128; off > 0; off >>= 1) {
        if (tid < off) {
            s0[tid] += s0[tid + off];
            s1[tid] += s1[tid + off];
            s2[tid] += s2[tid + off];
            s3[tid] += s3[tid + off];
        }
        __syncthreads();
    }
    if (tid == 0) {
        const float invBD  = 1.0f / ((float)BATCHN * (float)DIMN);
        const float invBDD = invBD / (float)DIMN;
        const float norms  = (s1[0] * invBD + s2[0] * invBD + s3[0] * invBDD) / 3.0f;
        loss_out[0] = s0[0] / (float)BATCHN + ALPHAF * norms;
    }
}

extern "C" void kernel_launch(void* const* d_in, const int* in_sizes, int n_in,
                              void* d_out, int out_size, void* d_ws, size_t ws_size,
                              hipStream_t stream) {
    (void)in_sizes; (void)n_in; (void)out_size; (void)ws_size;
    const int*   h_idx  = (const int*)d_in[0];
    const int*   r_idx  = (const int*)d_in[1];
    const int*   t_idx  = (const int*)d_in[2];
    const float* labels = (const float*)d_in[3];
    const float* ent_w  = (const float*)d_in[4];
    const float* rel_w  = (const float*)d_in[5];

    float* out      = (float*)d_out;
    float* loss_out = out;          // d_out[0]
    float* scores   = out + 1;      // d_out[1..BATCHN]
    float* partials = (float*)d_ws; // BATCHN*4 floats = 256 KB scratch

    rescal_score_kernel<<<BATCHN, 128, 0, stream>>>(
        h_idx, r_idx, t_idx, labels, ent_w, rel_w, scores, partials);
    rescal_loss_kernel<<<1, 256, 0, stream>>>(partials, loss_out);
}